// ATTGNN_5892695130184
// MI455X (gfx1250) — compile-verified
//
#include <hip/hip_runtime.h>
#include <hip/hip_bf16.h>

typedef float v2f __attribute__((ext_vector_type(2)));
typedef float v8f __attribute__((ext_vector_type(8)));

#define N_NODES   100000
#define N_EDGES   1600000
#define IN_DIM    128
#define HDIM      64
#define N_CLASSES 16
#define NEG_SLOPE 0.2f

// ---------------------------------------------------------------------------
// Dense GEMM  C[M x N] = A[M x K] @ B[K x N] (+bias) using fp32 WMMA 16x16x4.
// One 256-thread block = 8 waves; each wave owns a 16-row tile; block = 128 rows.
// B (small) is staged in LDS.
// A-fragment per ISA 7.12.2: lanes 0-15 hold K=+0,+1 ; lanes 16-31 hold K=+2,+3.
// B/C/D: one row striped across lanes within a VGPR.
// ---------------------------------------------------------------------------
template <int K, int N>
__global__ __launch_bounds__(256) void gemm_wmma_f32(
    const float* __restrict__ A, const float* __restrict__ B,
    const float* __restrict__ bias, float* __restrict__ C, int M) {
  __shared__ float Bs[K * N];
  const int tid = threadIdx.x;
  for (int i = tid; i < K * N; i += 256) Bs[i] = B[i];
  __syncthreads();

  const int wave = tid >> 5;
  const int lane = tid & 31;
  const int hi = lane >> 4;   // half-wave select
  const int lo = lane & 15;

  const long m0 = (long)blockIdx.x * 128 + wave * 16;
  long r = m0 + lo;
  if (r > M - 1) r = M - 1;           // clamp loads; stores are guarded
  const float* __restrict__ Arow = A + (size_t)r * K;

  constexpr int NT = N / 16;
  v8f acc[NT] = {};

  for (int kk = 0; kk < K / 4; ++kk) {
    const int kb = kk * 4 + 2 * hi;
    v2f a;
    a.x = Arow[kb];
    a.y = Arow[kb + 1];
#pragma unroll
    for (int nt = 0; nt < NT; ++nt) {
      v2f b;
      b.x = Bs[kb * N + nt * 16 + lo];
      b.y = Bs[(kb + 1) * N + nt * 16 + lo];
      acc[nt] = __builtin_amdgcn_wmma_f32_16x16x4_f32(
          /*neg_a=*/false, a, /*neg_b=*/false, b,
          /*c_mod=*/(short)0, acc[nt], /*reuse_a=*/false, /*reuse_b=*/false);
    }
  }

#pragma unroll
  for (int nt = 0; nt < NT; ++nt) {
    const int col = nt * 16 + lo;
    const float bv = bias ? bias[col] : 0.0f;
#pragma unroll
    for (int rr = 0; rr < 8; ++rr) {
      const long row = m0 + rr + 8 * hi;   // VGPR rr: M=rr (lo half), M=rr+8 (hi half)
      if (row < M) C[(size_t)row * N + col] = acc[nt][rr] + bv;
    }
  }
}

// ---------------------------------------------------------------------------
// Per-node: alpha_s[n]=h[n].a_src, alpha_d[n]=h[n].a_dst, m[n]=lrelu(as+ad)
// (self-loop logit seeds the segment-max). One wave32 per node, 64 features.
// ---------------------------------------------------------------------------
__global__ __launch_bounds__(256) void node_prep(
    const float* __restrict__ h, const float* __restrict__ a_src,
    const float* __restrict__ a_dst, float* __restrict__ alpha_s,
    float* __restrict__ alpha_d, float* __restrict__ m, int n_nodes) {
  const int wave = threadIdx.x >> 5;
  const int lane = threadIdx.x & 31;
  const int n = blockIdx.x * 8 + wave;
  if (n >= n_nodes) return;
  const float* hr = h + (size_t)n * HDIM;
  const float h0 = hr[lane], h1 = hr[lane + 32];
  float ps = h0 * a_src[lane] + h1 * a_src[lane + 32];
  float pd = h0 * a_dst[lane] + h1 * a_dst[lane + 32];
#pragma unroll
  for (int off = 16; off > 0; off >>= 1) {
    ps += __shfl_xor(ps, off, 32);
    pd += __shfl_xor(pd, off, 32);
  }
  if (lane == 0) {
    alpha_s[n] = ps;
    alpha_d[n] = pd;
    float e = ps + pd;
    m[n] = (e > 0.0f) ? e : NEG_SLOPE * e;   // self-loop seeds max
  }
}

// Per-edge: e = lrelu(as[src]+ad[dst]); segment max via native f32 atomic max.
__global__ __launch_bounds__(256) void edge_max(
    const int* __restrict__ adj, const float* __restrict__ alpha_s,
    const float* __restrict__ alpha_d, float* __restrict__ ebuf,
    float* __restrict__ m, int n_edges) {
  const int i = blockIdx.x * 256 + threadIdx.x;
  if (i >= n_edges) return;
  const int src = adj[i];
  const int dst = adj[n_edges + i];
  float e = alpha_s[src] + alpha_d[dst];
  e = (e > 0.0f) ? e : NEG_SLOPE * e;
  ebuf[i] = e;
  atomicMax(&m[dst], e);   // global_atomic_max_num_f32
}

// Per-node after max: p_self = exp(e_self - m); s = p_self; acc = h * p_self.
__global__ __launch_bounds__(256) void node_post(
    const float* __restrict__ h, const float* __restrict__ alpha_s,
    const float* __restrict__ alpha_d, const float* __restrict__ m,
    float* __restrict__ s, float* __restrict__ acc, int n_nodes) {
  const int wave = threadIdx.x >> 5;
  const int lane = threadIdx.x & 31;
  const int n = blockIdx.x * 8 + wave;
  if (n >= n_nodes) return;
  float e = alpha_s[n] + alpha_d[n];
  e = (e > 0.0f) ? e : NEG_SLOPE * e;
  const float pself = __expf(e - m[n]);
  if (lane == 0) s[n] = pself;
  const size_t base = (size_t)n * HDIM;
  acc[base + lane]      = h[base + lane] * pself;
  acc[base + lane + 32] = h[base + lane + 32] * pself;
}

// Per-edge: p = exp(e - m[dst]); s[dst] += p (native f32 atomic add); keep p.
__global__ __launch_bounds__(256) void edge_expsum(
    const int* __restrict__ adj, float* __restrict__ ebuf,
    const float* __restrict__ m, float* __restrict__ s, int n_edges) {
  const int i = blockIdx.x * 256 + threadIdx.x;
  if (i >= n_edges) return;
  const int dst = adj[n_edges + i];
  const float p = __expf(ebuf[i] - m[dst]);
  ebuf[i] = p;
  atomicAdd(&s[dst], p);
}

// Per-edge (one wave32 per edge): acc[dst,:] += h[src,:] * p. 64 feats/edge.
__global__ __launch_bounds__(256) void edge_scatter(
    const int* __restrict__ adj, const float* __restrict__ ebuf,
    const float* __restrict__ h, float* __restrict__ acc, int n_edges) {
  const int wid = (blockIdx.x * 256 + threadIdx.x) >> 5;
  if (wid >= n_edges) return;
  const int lane = threadIdx.x & 31;
  const int src = adj[wid];
  const int dst = adj[n_edges + wid];
  const float w = ebuf[wid];
  const float* __restrict__ hs = h + (size_t)src * HDIM;
  float* __restrict__ ao = acc + (size_t)dst * HDIM;
  atomicAdd(&ao[lane],      hs[lane]      * w);
  atomicAdd(&ao[lane + 32], hs[lane + 32] * w);
}

// Per-element: out = acc / s[n] + b[c], optional ELU.
__global__ __launch_bounds__(256) void normalize_bias(
    const float* __restrict__ acc, const float* __restrict__ s,
    const float* __restrict__ b, float* __restrict__ out, int total,
    int apply_elu) {
  const int idx = blockIdx.x * 256 + threadIdx.x;
  if (idx >= total) return;
  const int n = idx >> 6;      // /HDIM
  const int c = idx & 63;
  float v = acc[idx] / s[n] + b[c];
  if (apply_elu) v = (v > 0.0f) ? v : (__expf(v) - 1.0f);
  out[idx] = v;
}

// ---------------------------------------------------------------------------
extern "C" void kernel_launch(void* const* d_in, const int* in_sizes, int n_in,
                              void* d_out, int out_size, void* d_ws,
                              size_t ws_size, hipStream_t stream) {
  const float* X      = (const float*)d_in[0];
  const int*   adj    = (const int*)d_in[1];   // [2, E] int32
  const float* W1     = (const float*)d_in[2];
  const float* a_src1 = (const float*)d_in[3];
  const float* a_dst1 = (const float*)d_in[4];
  const float* b1     = (const float*)d_in[5];
  const float* W2     = (const float*)d_in[6];
  const float* a_src2 = (const float*)d_in[7];
  const float* a_dst2 = (const float*)d_in[8];
  const float* b2     = (const float*)d_in[9];
  const float* clas_W = (const float*)d_in[10];
  const float* clas_b = (const float*)d_in[11];

  float* out = (float*)d_out;                   // [logits 100000x16 | x 100000x64]
  float* x2  = out + (size_t)N_NODES * N_CLASSES;

  float* ws   = (float*)d_ws;
  float* h    = ws;                                   // 100000*64
  float* x1   = ws + (size_t)N_NODES * HDIM;          // 100000*64
  float* acc  = ws + (size_t)2 * N_NODES * HDIM;      // 100000*64
  float* as_  = ws + (size_t)3 * N_NODES * HDIM;      // 100000
  float* ad_  = as_ + N_NODES;                        // 100000
  float* m    = ad_ + N_NODES;                        // 100000
  float* s    = m + N_NODES;                          // 100000
  float* ebuf = s + N_NODES;                          // 1600000

  const int gemm_blocks = (N_NODES + 127) / 128;      // 782
  const int node_blocks = (N_NODES + 7) / 8;          // 12500
  const int edge_blocks = (N_EDGES + 255) / 256;      // 6250
  const int scat_blocks = (N_EDGES + 7) / 8;          // 200000
  const int norm_total  = N_NODES * HDIM;
  const int norm_blocks = (norm_total + 255) / 256;

  // ---- Layer 1 ----
  gemm_wmma_f32<IN_DIM, HDIM><<<gemm_blocks, 256, 0, stream>>>(X, W1, nullptr, h, N_NODES);
  node_prep<<<node_blocks, 256, 0, stream>>>(h, a_src1, a_dst1, as_, ad_, m, N_NODES);
  edge_max<<<edge_blocks, 256, 0, stream>>>(adj, as_, ad_, ebuf, m, N_EDGES);
  node_post<<<node_blocks, 256, 0, stream>>>(h, as_, ad_, m, s, acc, N_NODES);
  edge_expsum<<<edge_blocks, 256, 0, stream>>>(adj, ebuf, m, s, N_EDGES);
  edge_scatter<<<scat_blocks, 256, 0, stream>>>(adj, ebuf, h, acc, N_EDGES);
  normalize_bias<<<norm_blocks, 256, 0, stream>>>(acc, s, b1, x1, norm_total, /*elu=*/1);

  // ---- Layer 2 ----
  gemm_wmma_f32<HDIM, HDIM><<<gemm_blocks, 256, 0, stream>>>(x1, W2, nullptr, h, N_NODES);
  node_prep<<<node_blocks, 256, 0, stream>>>(h, a_src2, a_dst2, as_, ad_, m, N_NODES);
  edge_max<<<edge_blocks, 256, 0, stream>>>(adj, as_, ad_, ebuf, m, N_EDGES);
  node_post<<<node_blocks, 256, 0, stream>>>(h, as_, ad_, m, s, acc, N_NODES);
  edge_expsum<<<edge_blocks, 256, 0, stream>>>(adj, ebuf, m, s, N_EDGES);
  edge_scatter<<<scat_blocks, 256, 0, stream>>>(adj, ebuf, h, acc, N_EDGES);
  normalize_bias<<<norm_blocks, 256, 0, stream>>>(acc, s, b2, x2, norm_total, /*elu=*/0);

  // ---- Classifier: logits = x2 @ clas_W + clas_b ----
  gemm_wmma_f32<HDIM, N_CLASSES><<<gemm_blocks, 256, 0, stream>>>(x2, clas_W, clas_b, out, N_NODES);
}